// RelationNonLocal_46102178955357
// MI455X (gfx1250) — compile-verified
//
#include <hip/hip_runtime.h>

// RelationNonLocal, algebraically collapsed (no softmax => attention is linear):
//   out = ( Q @ (K^T V) / HW ) @ W3^T + b3, with Q/K/V = 1x1-conv projections.
// All matmuls run on V_WMMA_F32_16X16X4_F32 (full f32, f32 accumulate).
// Bias folded into WMMA C-operand init (bias depends only on N=lane index).

#define C_CH 256
#define HW_P 4096
#define N_B  4
#define PCHUNKS 8              // split of the 4096-deep Gram reduction
#define PCHUNK_LEN (HW_P / PCHUNKS)

typedef __attribute__((ext_vector_type(2))) float v2f;
typedef __attribute__((ext_vector_type(4))) float v4f;
typedef __attribute__((ext_vector_type(8))) float v8f;

__device__ __forceinline__ v8f wmma_f32(v2f a, v2f b, v8f c) {
  // 8 args: (neg_a, A, neg_b, B, c_mod, C, reuse_a, reuse_b)
  return __builtin_amdgcn_wmma_f32_16x16x4_f32(false, a, false, b, (short)0, c,
                                               false, false);
}

__device__ __forceinline__ v8f splat8(float v) {
  v8f r = {v, v, v, v, v, v, v, v};
  return r;
}

// ---------------------------------------------------------------------------
// Kernel 1: V/K/Q projections. Y[n,p,o] = sum_c x[n,c,p] * w[o,c] + b[o]
// One wave -> 16(pixels) x 32(out-ch): two WMMA tiles sharing the A operand.
// ---------------------------------------------------------------------------
__global__ void __launch_bounds__(256)
qkv_proj_kernel(const float* __restrict__ x,
                const float* __restrict__ w0, const float* __restrict__ b0,
                const float* __restrict__ w1, const float* __restrict__ b1,
                const float* __restrict__ w2, const float* __restrict__ b2,
                float* __restrict__ Vb, float* __restrict__ Kb,
                float* __restrict__ Qb) {
  const int lane = threadIdx.x & 31;
  const int l15  = lane & 15;
  const int hi   = lane >> 4;
  int t = blockIdx.x * (blockDim.x >> 5) + (threadIdx.x >> 5);
  const int opair = t & 7;   t >>= 3;   // 8 pairs of 16 out-channels
  const int ptile = t & 255; t >>= 8;   // 256 pixel tiles
  const int n     = t & 3;   t >>= 2;
  const int proj  = t;                  // 0..2  (wave-uniform)

  const float* w    = (proj == 0) ? w0 : (proj == 1) ? w1 : w2;
  const float* bias = (proj == 0) ? b0 : (proj == 1) ? b1 : b2;
  float* out        = (proj == 0) ? Vb : (proj == 1) ? Kb : Qb;

  const int p0 = ptile * 16;
  const int o0 = opair * 32;
  const float* xn    = x + (size_t)n * C_CH * HW_P;
  const float* wrow0 = w + (size_t)(o0 + l15) * C_CH;       // B0 lane row
  const float* wrow1 = w + (size_t)(o0 + 16 + l15) * C_CH;  // B1 lane row

  v8f acc0 = splat8(bias[o0 + l15]);        // bias as WMMA C-input
  v8f acc1 = splat8(bias[o0 + 16 + l15]);
#pragma unroll 4
  for (int k0 = 0; k0 < C_CH; k0 += 4) {
    const int cb = k0 + 2 * hi;            // wave-half selects K pair
    v2f a;
    a.x = xn[(size_t)cb * HW_P + p0 + l15];        // A[m=l15][k]
    a.y = xn[(size_t)(cb + 1) * HW_P + p0 + l15];
    v2f bb0 = *(const v2f*)(wrow0 + cb);           // B[k][n=l15]
    v2f bb1 = *(const v2f*)(wrow1 + cb);
    acc0 = wmma_f32(a, bb0, acc0);
    acc1 = wmma_f32(a, bb1, acc1);
  }
  float* obase = out + ((size_t)n * HW_P + p0) * C_CH + o0 + l15;
#pragma unroll
  for (int r = 0; r < 8; ++r) {
    obase[(size_t)(r + 8 * hi) * C_CH]      = acc0[r];  // D: M=r+8*hi, N=l15
    obase[(size_t)(r + 8 * hi) * C_CH + 16] = acc1[r];
  }
}

// ---------------------------------------------------------------------------
// Kernel 2: partial Gram. Mpart[ch,n,d,c] = sum_{p in chunk} K[n,p,d]*V[n,p,c]
// 8 p-chunks of 512 -> 8192 waves of 128 WMMA steps each.
// ---------------------------------------------------------------------------
__global__ void __launch_bounds__(256)
kv_gram_kernel(const float* __restrict__ Kb, const float* __restrict__ Vb,
               float* __restrict__ Mpart) {
  const int lane = threadIdx.x & 31;
  const int l15  = lane & 15;
  const int hi   = lane >> 4;
  int t = blockIdx.x * (blockDim.x >> 5) + (threadIdx.x >> 5);
  const int chunk = t & 7;  t >>= 3;
  const int ctile = t & 15; t >>= 4;
  const int dtile = t & 15; t >>= 4;
  const int n     = t;

  const int d0 = dtile * 16, c0 = ctile * 16;
  const int pbeg = chunk * PCHUNK_LEN;
  const float* kbase = Kb + (size_t)n * HW_P * C_CH;
  const float* vbase = Vb + (size_t)n * HW_P * C_CH;

  v8f acc = {};
#pragma unroll 4
  for (int p = pbeg; p < pbeg + PCHUNK_LEN; p += 4) {
    const int pb = p + 2 * hi;
    v2f a, b;
    a.x = kbase[(size_t)pb * C_CH + d0 + l15];        // A[m=d][k=p]
    a.y = kbase[(size_t)(pb + 1) * C_CH + d0 + l15];
    b.x = vbase[(size_t)pb * C_CH + c0 + l15];        // B[k=p][n=c]
    b.y = vbase[(size_t)(pb + 1) * C_CH + c0 + l15];
    acc = wmma_f32(a, b, acc);
  }
  float* mp = Mpart + ((size_t)chunk * N_B + n) * C_CH * C_CH +
              (size_t)d0 * C_CH + c0 + l15;
#pragma unroll
  for (int r = 0; r < 8; ++r)
    mp[(size_t)(r + 8 * hi) * C_CH] = acc[r];
}

// ---------------------------------------------------------------------------
// Kernel 2b: deterministic sequential sum of the 8 partial Gram slabs.
// ---------------------------------------------------------------------------
__global__ void __launch_bounds__(256)
reduce_m_kernel(const float* __restrict__ Mpart, float* __restrict__ Mb) {
  const size_t idx    = (size_t)blockIdx.x * blockDim.x + threadIdx.x;
  const size_t stride = (size_t)N_B * C_CH * C_CH;
  float s = 0.0f;
#pragma unroll
  for (int c = 0; c < PCHUNKS; ++c) s += Mpart[c * stride + idx];
  Mb[idx] = s;
}

// ---------------------------------------------------------------------------
// Kernel 3: M2[n,d,o] = (sum_c M[n,d,c] * w3[o,c]) / HW  (folds conv3 weight)
// ---------------------------------------------------------------------------
__global__ void __launch_bounds__(256)
m_w3_kernel(const float* __restrict__ Mb, const float* __restrict__ w3,
            float* __restrict__ M2) {
  const int lane = threadIdx.x & 31;
  const int l15  = lane & 15;
  const int hi   = lane >> 4;
  int t = blockIdx.x * (blockDim.x >> 5) + (threadIdx.x >> 5);
  const int otile = t & 15; t >>= 4;
  const int dtile = t & 15; t >>= 4;
  const int n     = t;
  const int d0 = dtile * 16, o0 = otile * 16;

  const float* mrow = Mb + ((size_t)n * C_CH + d0 + l15) * C_CH;
  const float* wrow = w3 + (size_t)(o0 + l15) * C_CH;
  v8f acc = {};
#pragma unroll 4
  for (int k0 = 0; k0 < C_CH; k0 += 4) {
    const int cb = k0 + 2 * hi;
    v2f a = *(const v2f*)(mrow + cb);
    v2f b = *(const v2f*)(wrow + cb);
    acc = wmma_f32(a, b, acc);
  }
  const float inv_hw = 1.0f / (float)HW_P;
  float* m2base = M2 + ((size_t)n * C_CH + d0) * C_CH + o0 + l15;
#pragma unroll
  for (int r = 0; r < 8; ++r)
    m2base[(size_t)(r + 8 * hi) * C_CH] = acc[r] * inv_hw;
}

// ---------------------------------------------------------------------------
// Kernel 4: out[n,o,p] = sum_d Q[n,p,d] * M2[n,d,o] + b3[o]   (NCHW store)
// One wave -> 16(pixels) x 32(out-ch): two WMMA tiles sharing the A operand.
// ---------------------------------------------------------------------------
__global__ void __launch_bounds__(256)
out_kernel(const float* __restrict__ Qb, const float* __restrict__ M2,
           const float* __restrict__ b3, float* __restrict__ out) {
  const int lane = threadIdx.x & 31;
  const int l15  = lane & 15;
  const int hi   = lane >> 4;
  int t = blockIdx.x * (blockDim.x >> 5) + (threadIdx.x >> 5);
  const int opair = t & 7;   t >>= 3;
  const int ptile = t & 255; t >>= 8;
  const int n     = t;
  const int p0 = ptile * 16, o0 = opair * 32;

  const float* qrow = Qb + ((size_t)n * HW_P + p0 + l15) * C_CH;
  const float* m2b  = M2 + (size_t)n * C_CH * C_CH;

  v8f acc0 = splat8(b3[o0 + l15]);        // bias as WMMA C-input
  v8f acc1 = splat8(b3[o0 + 16 + l15]);
#pragma unroll 4
  for (int k0 = 0; k0 < C_CH; k0 += 4) {
    const int db = k0 + 2 * hi;
    v2f a = *(const v2f*)(qrow + db);                 // A[m=pixel][k=d]
    v2f bb0, bb1;
    bb0.x = m2b[(size_t)db * C_CH + o0 + l15];        // B[k=d][n=o]
    bb0.y = m2b[(size_t)(db + 1) * C_CH + o0 + l15];
    bb1.x = m2b[(size_t)db * C_CH + o0 + 16 + l15];
    bb1.y = m2b[(size_t)(db + 1) * C_CH + o0 + 16 + l15];
    acc0 = wmma_f32(a, bb0, acc0);
    acc1 = wmma_f32(a, bb1, acc1);
  }
  // lane's results share channel o, cover 8 consecutive pixels -> b128 stores
  float* ob0 = out + ((size_t)n * C_CH + o0 + l15) * HW_P + p0 + 8 * hi;
  float* ob1 = out + ((size_t)n * C_CH + o0 + 16 + l15) * HW_P + p0 + 8 * hi;
  v4f a_lo = {acc0[0], acc0[1], acc0[2], acc0[3]};
  v4f a_hi = {acc0[4], acc0[5], acc0[6], acc0[7]};
  v4f b_lo = {acc1[0], acc1[1], acc1[2], acc1[3]};
  v4f b_hi = {acc1[4], acc1[5], acc1[6], acc1[7]};
  *(v4f*)(ob0)     = a_lo;
  *(v4f*)(ob0 + 4) = a_hi;
  *(v4f*)(ob1)     = b_lo;
  *(v4f*)(ob1 + 4) = b_hi;
}

extern "C" void kernel_launch(void* const* d_in, const int* in_sizes, int n_in,
                              void* d_out, int out_size, void* d_ws,
                              size_t ws_size, hipStream_t stream) {
  const float* x  = (const float*)d_in[0];
  const float* w0 = (const float*)d_in[1];
  const float* b0 = (const float*)d_in[2];
  const float* w1 = (const float*)d_in[3];
  const float* b1 = (const float*)d_in[4];
  const float* w2 = (const float*)d_in[5];
  const float* b2 = (const float*)d_in[6];
  const float* w3 = (const float*)d_in[7];
  const float* b3 = (const float*)d_in[8];
  float* out = (float*)d_out;

  const size_t per = (size_t)N_B * HW_P * C_CH;   // 4 Mi floats each
  const size_t mcc = (size_t)N_B * C_CH * C_CH;   // 256 Ki floats
  float* Vb    = (float*)d_ws;
  float* Kb    = Vb + per;
  float* Qb    = Kb + per;
  float* Mpart = Qb + per;            // [PCHUNKS, N, C, C]
  float* Mb    = Mpart + PCHUNKS * mcc;
  float* M2    = Mb + mcc;

  // waves: 3*4*256*8 = 24576 -> 3072 blocks of 8 waves
  qkv_proj_kernel<<<3072, 256, 0, stream>>>(x, w0, b0, w1, b1, w2, b2, Vb, Kb, Qb);
  // waves: 4*16*16*8 = 8192 -> 1024 blocks
  kv_gram_kernel<<<1024, 256, 0, stream>>>(Kb, Vb, Mpart);
  // 262144 elems -> 1024 blocks of 256 threads
  reduce_m_kernel<<<1024, 256, 0, stream>>>(Mpart, Mb);
  // waves: 4*16*16 = 1024 -> 128 blocks
  m_w3_kernel<<<128, 256, 0, stream>>>(Mb, w3, M2);
  // waves: 4*256*8 = 8192 -> 1024 blocks
  out_kernel<<<1024, 256, 0, stream>>>(Qb, M2, b3, out);
}